// SelfOrganizingMap_20658792694116
// MI455X (gfx1250) — compile-verified
//
#include <hip/hip_runtime.h>

// ---------------------------------------------------------------------------
// Persistent cluster-resident SOM trainer for MI455X (gfx1250).
//   grid = 16 workgroups (one 16-WG cluster, one WGP each), block = 256 (8 waves)
//   - 64x384 f32 weight slab per WG lives in LDS for the whole kernel (96 KB).
//   - w·x via chained V_WMMA_F32_16X16X4_F32 (f32 precision for the argmin).
//   - BMU: packed (sortable-f32-key<<32 | index) u64 atomicMin in L2.
//   - Cluster-wide sync: s_barrier_signal/-wait -3 (two per step).
//   - Sample vector double-buffered in LDS, prefetched one step ahead by the
//     Tensor Data Mover (tensor_load_to_lds + s_wait_tensorcnt).
// ---------------------------------------------------------------------------

typedef __attribute__((ext_vector_type(2))) float v2f;
typedef __attribute__((ext_vector_type(8))) float v8f;
typedef __attribute__((ext_vector_type(4))) unsigned int v4u;
typedef __attribute__((ext_vector_type(8))) int v8i_;
typedef __attribute__((ext_vector_type(4))) int v4i_;

#define SOM_LR        0.1f
#define SOM_INVDENOM  0.5f      // 1/(2*sigma^2), sigma = 1
#define N_SAMPLES     8192
#define DIM           384
#define CELLS_PER_WG  64
#define NWG           16

__device__ __forceinline__ void cluster_barrier_all(unsigned tid) {
    // WG-sync, one wave signals the cluster barrier (-3), all waves wait.
    __syncthreads();
    if (tid < 32) asm volatile("s_barrier_signal -3" ::: "memory");
    asm volatile("s_barrier_wait -3" ::: "memory");
}

// TDM: 1-D tile load of DIM f32 from global -> LDS (issued once per wave).
// D# group0: count=1 | lds_addr | global_addr[56:0] | type=2.
// D# group1: data_size=4B, tensor_dim0=tile_dim0=384, tensor_dim1=tile_dim1=1,
//            tensor_dim0_stride=384. Groups 2/3 zero (<=2D tensor).
__device__ __forceinline__ void tdm_load_x(const float* gsrc, unsigned lds_addr) {
    const unsigned long long ga = (unsigned long long)(uintptr_t)gsrc;
    v4u g0;
    g0.x = 0x1u;                                            // count=1, user D#
    g0.y = lds_addr;                                        // LDS byte address
    g0.z = (unsigned)(ga & 0xFFFFFFFFu);                    // global_addr[31:0]
    g0.w = (unsigned)((ga >> 32) & 0x01FFFFFFu) | (2u << 30); // [56:32] | type=2
    v8i_ g1;
    g1[0] = 0x00020000;                 // workgroup_mask=0, data_size=4B(code 2)
    g1[1] = (int)(384u << 16);          // tensor_dim0[15:0] = 384
    g1[2] = (int)(1u << 16);            // tensor_dim0 hi=0 | tensor_dim1 lo=1
    g1[3] = (int)(384u << 16);          // tensor_dim1 hi=0 | tile_dim0=384
    g1[4] = 1;                          // tile_dim1=1, tile_dim2=0
    g1[5] = 384;                        // tensor_dim0_stride lo
    g1[6] = 0;
    g1[7] = 0;                          // tensor_dim1_stride=0 (unused, 1 row)
    const v4i_ gz = {0, 0, 0, 0};
#if __clang_major__ >= 23
    const v8i_ gz8 = {0, 0, 0, 0, 0, 0, 0, 0};
    __builtin_amdgcn_tensor_load_to_lds(g0, g1, gz, gz, gz8, 0);
#else
    __builtin_amdgcn_tensor_load_to_lds(g0, g1, gz, gz, 0);
#endif
}

__global__ __launch_bounds__(256)
void SelfOrganizingMap_20658792694116_kernel(const float* __restrict__ x_all,
                                             const float* __restrict__ w_init,
                                             const int*   __restrict__ n_epochs_p,
                                             float*       __restrict__ out,
                                             unsigned long long* __restrict__ slots) {
    __shared__ __align__(16) float wlds[CELLS_PER_WG * DIM];   // 96 KB weight slab
    __shared__ __align__(16) float xbuf2[2][DIM];              // double-buffered x
    __shared__ float dots2[2][CELLS_PER_WG];                   // K-half partial dots
    __shared__ float dotsum[CELLS_PER_WG];
    __shared__ float normsq[CELLS_PER_WG];                     // incremental ||w||^2
    __shared__ unsigned long long lmin;                        // WG-local packed min
    __shared__ unsigned int bmuS;
    __shared__ float xnormS;

    const unsigned tid = threadIdx.x;
    const unsigned wg  = blockIdx.x;              // 0..15 = workgroup-in-cluster
    const long long total = (long long)n_epochs_p[0] * N_SAMPLES;

    // ---- init: weights -> LDS, seed ||w||^2, reset BMU slots, prefetch x0 --
    {
        const float* wsrc = w_init + (size_t)wg * CELLS_PER_WG * DIM;
        for (int i = tid * 4; i < CELLS_PER_WG * DIM; i += 256 * 4)
            *(float4*)&wlds[i] = *(const float4*)&wsrc[i];
        __syncthreads();
        if (tid < CELLS_PER_WG) {
            float s = 0.f;
            const float* wr = &wlds[tid * DIM];
            for (int k = 0; k < DIM; ++k) { float v = wr[k]; s += v * v; }
            normsq[tid] = s;
        }
        if (wg == 0 && tid == 0) {
            atomicExch(&slots[0], ~0ull);
            atomicExch(&slots[1], ~0ull);
        }
        if (tid < 32)   // wave 0: TDM-stage sample 0 into buffer 0
            tdm_load_x(x_all, (unsigned)(uintptr_t)&xbuf2[0][0]);
        __threadfence();
    }
    cluster_barrier_all(tid);

    for (long long t = 0; t < total; ++t) {
        const int cur = (int)(t & 1);             // x buffer + BMU slot parity

        // ---- wave 0: complete current x DMA, launch prefetch of next ------
        if (tid < 32) {
            __builtin_amdgcn_s_wait_tensorcnt(0);
            const long long nt = (t + 1 < total) ? t + 1 : t;   // clamped
            const int ns = (int)(nt & (N_SAMPLES - 1));
            tdm_load_x(x_all + (size_t)ns * DIM,
                       (unsigned)(uintptr_t)&xbuf2[cur ^ 1][0]);
        }
        if (tid == 0) lmin = ~0ull;
        __syncthreads();
        const float* xbuf = xbuf2[cur];

        // ---- w·x for all 64 cells via V_WMMA_F32_16X16X4_F32 --------------
        // wave wv: tile = wv/2 (16 cells), K-half = wv&1 (192 dims, 48 WMMAs).
        // A 16x4: lanes 0-15 hold {K0,K1}, lanes 16-31 hold {K2,K3};
        // B columns replicate the x chunk, so D[m][*] = partial dot(m).
        {
            const int lane = tid & 31;
            const int tile = (tid >> 5) >> 1;
            const int kh   = (tid >> 5) & 1;
            const int mrow = lane & 15;
            const int ksel = (lane >> 4) << 1;                 // 0 or 2
            const float* wr = &wlds[(tile * 16 + mrow) * DIM];
            const int kbase = kh * (DIM / 2);
            v8f acc = {};
            for (int kc = 0; kc < DIM / 2; kc += 4) {
                const int ko = kbase + kc + ksel;
                v2f a, b;
                a.x = wr[ko];   a.y = wr[ko + 1];
                b.x = xbuf[ko]; b.y = xbuf[ko + 1];
                acc = __builtin_amdgcn_wmma_f32_16x16x4_f32(
                          false, a, false, b, (short)0, acc, false, false);
            }
            if (mrow == 0) {                                   // lanes 0 & 16: N==0
                const int base = tile * 16 + ((lane >> 4) << 3);
#pragma unroll
                for (int r = 0; r < 8; ++r) dots2[kh][base + r] = acc[r];
            }
        }
        __syncthreads();

        // ---- score + WG-local packed argmin; wave 2 computes ||x||^2 ------
        if (tid < CELLS_PER_WG) {
            const float dot = dots2[0][tid] + dots2[1][tid];
            dotsum[tid] = dot;
            const float score = normsq[tid] - 2.f * dot;       // d^2 - ||x||^2
            unsigned int kb = __float_as_uint(score);
            kb = (kb & 0x80000000u) ? ~kb : (kb | 0x80000000u);// sortable-float map
            const unsigned long long pk =
                ((unsigned long long)kb << 32) | (unsigned)(wg * CELLS_PER_WG + tid);
            atomicMin(&lmin, pk);
        } else if (tid < 96) {
            const int l = tid - 64;
            float sx = 0.f;
            for (int j = l * 12; j < l * 12 + 12; ++j) { float xv = xbuf[j]; sx += xv * xv; }
            for (int off = 16; off > 0; off >>= 1) sx += __shfl_down(sx, off, 32);
            if (l == 0) xnormS = sx;
        }
        __syncthreads();

        // ---- cluster-wide BMU reduction in L2 -----------------------------
        if (tid == 0) atomicMin(&slots[cur], lmin);
        __threadfence();
        cluster_barrier_all(tid);

        if (tid == 0) {
            const unsigned long long got = atomicAdd(&slots[cur], 0ull); // coherent read
            bmuS = (unsigned int)got;
            if (wg == 0) atomicExch(&slots[cur ^ 1], ~0ull);   // recycle other slot
        }
        __syncthreads();

        // ---- neighborhood-weighted update of the LDS slab -----------------
        {
            const unsigned bmu = bmuS;
            const float bx = (float)(bmu >> 5), by = (float)(bmu & 31);
            const int c = tid >> 2, q = tid & 3;               // 4 threads per cell
            const int g = wg * CELLS_PER_WG + c;
            const float dx = (float)(g >> 5) - bx, dy = (float)(g & 31) - by;
            const float gd = sqrtf(dx * dx + dy * dy);
            const float a  = SOM_LR * __expf(-gd * SOM_INVDENOM);
            float* wr = &wlds[c * DIM + q * (DIM / 4)];
            const float* xr = &xbuf[q * (DIM / 4)];
#pragma unroll 4
            for (int k = 0; k < DIM / 4; k += 4) {
                float4 w4 = *(float4*)&wr[k];
                const float4 x4 = *(const float4*)&xr[k];
                w4.x += a * (x4.x - w4.x);  w4.y += a * (x4.y - w4.y);
                w4.z += a * (x4.z - w4.z);  w4.w += a * (x4.w - w4.w);
                *(float4*)&wr[k] = w4;
            }
            if (q == 0) {   // ||w'||^2 = (1-a)^2||w||^2 + 2a(1-a)(w·x) + a^2||x||^2
                const float om = 1.f - a;
                normsq[c] = om * om * normsq[c] + 2.f * a * om * dotsum[c]
                          + a * a * xnormS;
            }
        }
        __threadfence();
        cluster_barrier_all(tid);   // protects slot reset vs next step's atomics
    }

    // ---- write back final weights ----------------------------------------
    __syncthreads();
    float* dst = out + (size_t)wg * CELLS_PER_WG * DIM;
    for (int i = tid * 4; i < CELLS_PER_WG * DIM; i += 256 * 4)
        *(float4*)&dst[i] = *(const float4*)&wlds[i];
}

extern "C" void kernel_launch(void* const* d_in, const int* in_sizes, int n_in,
                              void* d_out, int out_size, void* d_ws, size_t ws_size,
                              hipStream_t stream) {
    (void)in_sizes; (void)n_in; (void)out_size; (void)ws_size;
    const float* x_all  = (const float*)d_in[0];   // [8192, 384] f32
    const float* w_init = (const float*)d_in[1];   // [32, 32, 384] f32
    // d_in[2] = grid_coordinates (recomputed analytically on device)
    const int*   ne     = (const int*)d_in[3];     // num_epochs (scalar)
    float* out = (float*)d_out;
    unsigned long long* slots = (unsigned long long*)d_ws;  // 2 x u64 BMU slots

    SelfOrganizingMap_20658792694116_kernel<<<dim3(NWG), dim3(256), 0, stream>>>(
        x_all, w_init, ne, out, slots);
}